// SS2D_31061203485229
// MI455X (gfx1250) — compile-verified
//
#include <hip/hip_runtime.h>
#include <hip/hip_bf16.h>
#include <math.h>

// ---------------------------------------------------------------------------
// SS2D (VMamba) forward for MI455X (gfx1250, wave32).
// GEMMs: V_WMMA_F32_16X16X4_F32, LDS-staged activation panel via
// GLOBAL_LOAD_ASYNC_TO_LDS_B128 (ASYNCcnt) when available.
// Selective scan: one state (n) per lane, 16 lanes/channel, 2 channels/wave32.
// ---------------------------------------------------------------------------

#define BSZ 4
#define DIMC 128
#define HH 64
#define WW 64
#define LL 4096          // H*W
#define DD 128
#define NN 16
#define RR 8
#define KK 4

typedef __attribute__((ext_vector_type(2))) float v2f;
typedef __attribute__((ext_vector_type(8))) float v8f;
typedef __attribute__((ext_vector_type(4))) int   v4i;

// address-space qualified vector pointers for the async-to-LDS builtin
typedef __attribute__((address_space(1))) v4i as1_v4i;   // global
typedef __attribute__((address_space(3))) v4i as3_v4i;   // LDS

#if defined(__has_builtin)
#if __has_builtin(__builtin_amdgcn_global_load_async_to_lds_b128) && \
    __has_builtin(__builtin_amdgcn_s_wait_asynccnt)
#define USE_ASYNC_LDS 1
#endif
#endif

__device__ __forceinline__ float gelu_exact(float x) {
    return 0.5f * x * (1.0f + erff(x * 0.70710678118654752440f));
}

// ---------------------------------------------------------------------------
// fp32 WMMA GEMM:  Out[z] = A[z/aDiv] (MxK, row-major) * X[z%4] (KxN)
// Block (256 thr, 8 waves) computes a 32(M) x 256(N) tile. K is consumed in
// 32-deep chunks staged into LDS (async-to-LDS path when the builtin exists).
// Each wave owns a 16x64 strip: one A fragment feeds 4 WMMAs from LDS.
// Requires Ncols % 256 == 0 and Kdim % 32 == 0 (both 4096/128 here).
// ---------------------------------------------------------------------------
__global__ __launch_bounds__(256) void gemm_f32_wmma(
    const float* __restrict__ A, const float* __restrict__ X,
    float* __restrict__ Out, int M, int Kdim, int Ncols,
    int aDiv, long long strideA, long long strideX, long long strideOut)
{
    __shared__ float xtile[32 * 256];                 // 32 KB K-chunk panel

    const int z = blockIdx.z;
    A   += (long long)(z / aDiv) * strideA;
    X   += (long long)(z % BSZ)  * strideX;
    Out += (long long)z * strideOut;

    const int t     = threadIdx.x;
    const int lane  = t & 31;
    const int wave  = t >> 5;
    const int lhalf = lane & 15;
    const int hi    = lane >> 4;                      // half-wave select

    const int nBase = blockIdx.x * 256;
    const int mtile = blockIdx.y * 2 + (wave >> 2);   // 16-row output tile
    const int nseg  = (wave & 3) * 64;                // 64-wide segment

    // A fragment row (clamped for partial M tiles; stores are guarded).
    int am = mtile * 16 + lhalf; if (am >= M) am = M - 1;
    const float* Arow = A + (long long)am * Kdim + 2 * hi;   // K = k0+kc+2*hi+{0,1}

    // Loader mapping: thread t stages row kc_row of the chunk, 8 float4 each.
    const int kc_row = t >> 3;                        // 0..31
    const int quad   = t & 7;                         // 0..7
    const float* gsrc = X + (long long)kc_row * Ncols + nBase + quad * 4;
    float*       ldst = &xtile[kc_row * 256 + quad * 4];

    v8f acc0 = {0.f,0.f,0.f,0.f,0.f,0.f,0.f,0.f};
    v8f acc1 = acc0, acc2 = acc0, acc3 = acc0;

    for (int k0 = 0; k0 < Kdim; k0 += 32) {
        __syncthreads();                               // panel reuse fence
#ifdef USE_ASYNC_LDS
#pragma unroll
        for (int i = 0; i < 8; ++i)
            __builtin_amdgcn_global_load_async_to_lds_b128(
                (as1_v4i*)(gsrc + (long long)k0 * Ncols + i * 32),
                (as3_v4i*)(ldst + i * 32),
                0, 0);
        __builtin_amdgcn_s_wait_asynccnt(0);
#else
#pragma unroll
        for (int i = 0; i < 8; ++i)
            *reinterpret_cast<float4*>(ldst + i * 32) =
                *reinterpret_cast<const float4*>(gsrc + (long long)k0 * Ncols + i * 32);
#endif
        __syncthreads();

#pragma unroll
        for (int kc = 0; kc < 32; kc += 4) {
            float2 av = *reinterpret_cast<const float2*>(Arow + k0 + kc);
            v2f a; a.x = av.x; a.y = av.y;

            const float* xr0 = &xtile[(kc + 2 * hi) * 256 + nseg + lhalf];
            const float* xr1 = xr0 + 256;
            v2f b0; b0.x = xr0[0];  b0.y = xr1[0];
            v2f b1; b1.x = xr0[16]; b1.y = xr1[16];
            v2f b2; b2.x = xr0[32]; b2.y = xr1[32];
            v2f b3; b3.x = xr0[48]; b3.y = xr1[48];

            acc0 = __builtin_amdgcn_wmma_f32_16x16x4_f32(false, a, false, b0, (short)0, acc0, false, false);
            acc1 = __builtin_amdgcn_wmma_f32_16x16x4_f32(false, a, false, b1, (short)0, acc1, false, false);
            acc2 = __builtin_amdgcn_wmma_f32_16x16x4_f32(false, a, false, b2, (short)0, acc2, false, false);
            acc3 = __builtin_amdgcn_wmma_f32_16x16x4_f32(false, a, false, b3, (short)0, acc3, false, false);
        }
    }

    const int mbase = mtile * 16 + hi * 8;             // VGPR r <-> row mbase+r
#pragma unroll
    for (int r = 0; r < 8; ++r) {
        int m = mbase + r;
        if (m < M) {
            long long o = (long long)m * Ncols + nBase + nseg + lhalf;
            Out[o]      = acc0[r];
            Out[o + 16] = acc1[r];
            Out[o + 32] = acc2[r];
            Out[o + 48] = acc3[r];
        }
    }
}

// ---------------------------------------------------------------------------
// Depthwise 3x3 conv (SAME, zero pad) + bias + GELU on xi half of xz,
// and z1 = GELU(z half). One thread per (b,d,h,w).
// ---------------------------------------------------------------------------
__global__ __launch_bounds__(256) void dwconv_gelu_kernel(
    const float* __restrict__ xz, const float* __restrict__ conv_w,
    const float* __restrict__ conv_b, float* __restrict__ z1,
    float* __restrict__ xi)
{
    int idx = blockIdx.x * 256 + threadIdx.x;          // B*D*L
    if (idx >= BSZ * DD * LL) return;
    int p = idx & (LL - 1);
    int d = (idx >> 12) & (DD - 1);
    int b = idx >> 19;
    int h = p >> 6, w = p & 63;

    const float* src = xz + ((long long)b * 2 * DD + d) * LL;
    float s = conv_b[d];
#pragma unroll
    for (int i = 0; i < 3; ++i) {
        int hh = h + i - 1;
        if ((unsigned)hh < (unsigned)HH) {
#pragma unroll
            for (int j = 0; j < 3; ++j) {
                int ww = w + j - 1;
                if ((unsigned)ww < (unsigned)WW)
                    s += src[hh * WW + ww] * conv_w[d * 9 + i * 3 + j];
            }
        }
    }
    xi[idx] = gelu_exact(s);

    float zz = xz[((long long)b * 2 * DD + DD + d) * LL + p];
    z1[idx] = gelu_exact(zz);
}

// Channel mean/max pool -> (b, 2, h, w)
__global__ __launch_bounds__(256) void pool_kernel(
    const float* __restrict__ xi, float* __restrict__ pooled)
{
    int idx = blockIdx.x * 256 + threadIdx.x;          // B*L
    if (idx >= BSZ * LL) return;
    int b = idx >> 12, p = idx & (LL - 1);
    const float* base = xi + (long long)b * DD * LL + p;
    float s = 0.f, mx = -3.4e38f;
    for (int d = 0; d < DD; ++d) {
        float v = base[(long long)d * LL];
        s += v; mx = fmaxf(mx, v);
    }
    pooled[((long long)b * 2 + 0) * LL + p] = s * (1.0f / DD);
    pooled[((long long)b * 2 + 1) * LL + p] = mx;
}

// 7x7 conv (2->1 ch, SAME) + bias + sigmoid -> attention map (b, h, w)
__global__ __launch_bounds__(256) void attn_kernel(
    const float* __restrict__ pooled, const float* __restrict__ sa_w,
    const float* __restrict__ sa_b, float* __restrict__ attn)
{
    int idx = blockIdx.x * 256 + threadIdx.x;          // B*L
    if (idx >= BSZ * LL) return;
    int b = idx >> 12, p = idx & (LL - 1);
    int h = p >> 6, w = p & 63;
    float acc = 0.f;
    for (int c = 0; c < 2; ++c) {
        const float* pc = pooled + ((long long)b * 2 + c) * LL;
        for (int i = 0; i < 7; ++i) {
            int hh = h + i - 3;
            if ((unsigned)hh >= (unsigned)HH) continue;
            for (int j = 0; j < 7; ++j) {
                int ww = w + j - 3;
                if ((unsigned)ww >= (unsigned)WW) continue;
                acc += pc[hh * WW + ww] * sa_w[c * 49 + i * 7 + j];
            }
        }
    }
    float x = acc + sa_b[0];
    attn[idx] = 1.0f / (1.0f + __expf(-x));
}

// xi *= attn (broadcast over channels)
__global__ __launch_bounds__(256) void scale_kernel(
    float* __restrict__ xi, const float* __restrict__ attn)
{
    int idx = blockIdx.x * 256 + threadIdx.x;          // B*D*L
    if (idx >= BSZ * DD * LL) return;
    int p = idx & (LL - 1);
    int b = idx >> 19;
    xi[idx] *= attn[(long long)b * LL + p];
}

// ---------------------------------------------------------------------------
// Selective scan. 16 lanes per channel (one per state n), 2 channels/wave32.
// dt-projection (K=8) + softplus fused per step. Directions are index
// permutations into the spatially-ordered projection buffer P and xi.
//   P layout: P[(k*BSZ+b)*40 + c][L], rows 0..7 = dt, 8..23 = B, 24..39 = C.
// ---------------------------------------------------------------------------
__global__ __launch_bounds__(256) void scan_kernel(
    const float* __restrict__ P, const float* __restrict__ XI,
    const float* __restrict__ dtw, const float* __restrict__ dtb,
    const float* __restrict__ A_logs, const float* __restrict__ Ds,
    float* __restrict__ YS)
{
    int tid = blockIdx.x * 256 + threadIdx.x;          // B*K*D*N lanes
    int n  = tid & (NN - 1);
    int ch = tid >> 4;                                 // 0..2047 = (b,k,d)
    int d  = ch & (DD - 1);
    int k  = (ch >> 7) & (KK - 1);
    int b  = ch >> 9;
    int kd = k * DD + d;

    const float Aval = -__expf(A_logs[kd * NN + n]);
    const float Dk   = Ds[kd];
    const float bias = dtb[kd];
    float w0 = dtw[kd * RR + 0], w1 = dtw[kd * RR + 1];
    float w2 = dtw[kd * RR + 2], w3 = dtw[kd * RR + 3];
    float w4 = dtw[kd * RR + 4], w5 = dtw[kd * RR + 5];
    float w6 = dtw[kd * RR + 6], w7 = dtw[kd * RR + 7];

    const float* Pkb    = P  + (long long)((k * BSZ + b) * 40) * LL;
    const float* u_base = XI + (long long)(b * DD + d) * LL;
    float*       ys     = YS + (long long)((b * KK + k) * DD + d) * LL;

    float h = 0.f;
    for (int l = 0; l < LL; ++l) {
        int ll  = (k >= 2) ? (LL - 1 - l) : l;
        int pos = (k & 1) ? (((ll & 63) << 6) | (ll >> 6)) : ll;  // transpose dirs

        // software prefetch ~8 steps ahead (global_prefetch_b8)
        int lf = l + 8;
        if (lf < LL) {
            int llf  = (k >= 2) ? (LL - 1 - lf) : lf;
            int posf = (k & 1) ? (((llf & 63) << 6) | (llf >> 6)) : llf;
            __builtin_prefetch(&u_base[posf], 0, 0);
            __builtin_prefetch(&Pkb[(RR + n) * LL + posf], 0, 0);
        }

        float u  = u_base[pos];
        float xr = bias;
        xr = fmaf(Pkb[0 * LL + pos], w0, xr);
        xr = fmaf(Pkb[1 * LL + pos], w1, xr);
        xr = fmaf(Pkb[2 * LL + pos], w2, xr);
        xr = fmaf(Pkb[3 * LL + pos], w3, xr);
        xr = fmaf(Pkb[4 * LL + pos], w4, xr);
        xr = fmaf(Pkb[5 * LL + pos], w5, xr);
        xr = fmaf(Pkb[6 * LL + pos], w6, xr);
        xr = fmaf(Pkb[7 * LL + pos], w7, xr);
        float dt = (xr > 20.f) ? xr : log1pf(__expf(xr));         // softplus

        float Bn = Pkb[(RR + n) * LL + pos];
        float Cn = Pkb[(RR + NN + n) * LL + pos];

        float dA = __expf(dt * Aval);
        h = fmaf(dA, h, dt * u * Bn);
        float part = h * Cn;

        // reduce over the 16 state lanes of this channel
        part += __shfl_xor(part, 8, 16);
        part += __shfl_xor(part, 4, 16);
        part += __shfl_xor(part, 2, 16);
        part += __shfl_xor(part, 1, 16);

        if (n == 0) ys[l] = part + u * Dk;
    }
}

// Cross-merge of the 4 directional scans (deterministic, no atomics).
__global__ __launch_bounds__(256) void merge_kernel(
    const float* __restrict__ YS, float* __restrict__ YM)
{
    int idx = blockIdx.x * 256 + threadIdx.x;          // B*D*L
    if (idx >= BSZ * DD * LL) return;
    int p = idx & (LL - 1);
    int d = (idx >> 12) & (DD - 1);
    int b = idx >> 19;
    int l1 = ((p & 63) << 6) | (p >> 6);               // inverse transpose index
    const float* base = YS + (long long)(b * KK) * DD * LL + (long long)d * LL;
    const long long kst = (long long)DD * LL;
    float v = base[0 * kst + p]
            + base[2 * kst + (LL - 1 - p)]
            + base[1 * kst + l1]
            + base[3 * kst + (LL - 1 - l1)];
    YM[idx] = v;
}

// Per-position channel LN (eps 1e-5) -> * z1 -> channel LN (eps 1e-6).
__global__ __launch_bounds__(256) void ln_gate_ln_kernel(
    const float* __restrict__ YM, const float* __restrict__ Z1,
    const float* __restrict__ g1, const float* __restrict__ b1,
    const float* __restrict__ g2, const float* __restrict__ b2,
    float* __restrict__ Y2)
{
    int idx = blockIdx.x * 256 + threadIdx.x;          // B*L positions
    if (idx >= BSZ * LL) return;
    int b = idx >> 12, p = idx & (LL - 1);
    const float* y = YM + (long long)b * DD * LL + p;
    const float* z = Z1 + (long long)b * DD * LL + p;
    float* o       = Y2 + (long long)b * DD * LL + p;

    float s = 0.f, ss = 0.f;
    for (int d = 0; d < DD; ++d) {
        float v = y[(long long)d * LL];
        s += v; ss += v * v;
    }
    float m1  = s * (1.0f / DD);
    float v1  = ss * (1.0f / DD) - m1 * m1;
    float rs1 = rsqrtf(v1 + 1e-5f);

    float s2 = 0.f, ss2 = 0.f;
    for (int d = 0; d < DD; ++d) {
        float tv = ((y[(long long)d * LL] - m1) * rs1 * g1[d] + b1[d]) * z[(long long)d * LL];
        s2 += tv; ss2 += tv * tv;
    }
    float m2  = s2 * (1.0f / DD);
    float v2  = ss2 * (1.0f / DD) - m2 * m2;
    float rs2 = rsqrtf(v2 + 1e-6f);

    for (int d = 0; d < DD; ++d) {
        float tv = ((y[(long long)d * LL] - m1) * rs1 * g1[d] + b1[d]) * z[(long long)d * LL];
        o[(long long)d * LL] = (tv - m2) * rs2 * g2[d] + b2[d];
    }
}

// ---------------------------------------------------------------------------
extern "C" void kernel_launch(void* const* d_in, const int* in_sizes, int n_in,
                              void* d_out, int out_size, void* d_ws, size_t ws_size,
                              hipStream_t stream)
{
    const float* x    = (const float*)d_in[0];
    const float* Wio  = (const float*)d_in[1];   // (256,128)
    const float* cw   = (const float*)d_in[2];   // (128,1,3,3)
    const float* cb   = (const float*)d_in[3];
    const float* saw  = (const float*)d_in[4];   // (1,2,7,7)
    const float* sab  = (const float*)d_in[5];
    const float* Wp   = (const float*)d_in[6];   // (4,40,128)
    const float* Wdt  = (const float*)d_in[7];   // (4,128,8)
    const float* bdt  = (const float*)d_in[8];   // (4,128)
    const float* Alog = (const float*)d_in[9];   // (512,16)
    const float* Dsk  = (const float*)d_in[10];  // (512,)
    const float* ong  = (const float*)d_in[11];
    const float* onb  = (const float*)d_in[12];
    const float* n1g  = (const float*)d_in[13];
    const float* n1b  = (const float*)d_in[14];
    const float* Wop  = (const float*)d_in[15];  // (128,128)
    float* out = (float*)d_out;

    // workspace layout (floats)
    float* ws = (float*)d_ws;
    long long off = 0;
    float* XZ     = ws + off; off += (long long)BSZ * 2 * DD * LL;   // 16.8 MB
    float* Z1     = ws + off; off += (long long)BSZ * DD * LL;       //  8.4 MB
    float* XI     = ws + off; off += (long long)BSZ * DD * LL;       //  8.4 MB
    float* POOLED = ws + off; off += (long long)BSZ * 2 * LL;
    float* ATTN   = ws + off; off += (long long)BSZ * LL;
    float* P      = ws + off; off += (long long)KK * BSZ * 40 * LL;  // 10.5 MB
    float* YS     = ws + off; off += (long long)BSZ * KK * DD * LL;  // 33.6 MB
    float* YM     = ws + off; off += (long long)BSZ * DD * LL;
    float* Y2     = ws + off; off += (long long)BSZ * DD * LL;

    const int ELTS = BSZ * DD * LL;       // 2,097,152
    const int POSB = BSZ * LL;            // 16,384
    const dim3 blk(256);

    // 1) in_proj: XZ[b] = Wio(256x128) * x[b](128x4096)
    gemm_f32_wmma<<<dim3(LL / 256, 2 * DD / 32, BSZ), blk, 0, stream>>>(
        Wio, x, XZ, 2 * DD, DIMC, LL,
        /*aDiv=*/1, /*strideA=*/0, (long long)DIMC * LL, (long long)2 * DD * LL);

    // 2) depthwise conv + GELU, and z1 = GELU(z)
    dwconv_gelu_kernel<<<dim3(ELTS / 256), blk, 0, stream>>>(XZ, cw, cb, Z1, XI);

    // 3) channel pool -> 7x7 attn conv + sigmoid -> gate xi
    pool_kernel<<<dim3(POSB / 256), blk, 0, stream>>>(XI, POOLED);
    attn_kernel<<<dim3(POSB / 256), blk, 0, stream>>>(POOLED, saw, sab, ATTN);
    scale_kernel<<<dim3(ELTS / 256), blk, 0, stream>>>(XI, ATTN);

    // 4) x_dbl projections, spatial order: P[k,b] = Wp[k](40x128) * XI[b]
    //    grid.y = 2 covers M tiles 0..3 (rows 0..63; rows >=40 guarded)
    gemm_f32_wmma<<<dim3(LL / 256, 2, KK * BSZ), blk, 0, stream>>>(
        Wp, XI, P, 40, DD, LL,
        /*aDiv=*/BSZ, (long long)40 * DD, (long long)DD * LL, (long long)40 * LL);

    // 5) selective scan, all 4 directions (dt proj + softplus fused)
    scan_kernel<<<dim3(BSZ * KK * DD * NN / 256), blk, 0, stream>>>(
        P, XI, Wdt, bdt, Alog, Dsk, YS);

    // 6) cross-merge
    merge_kernel<<<dim3(ELTS / 256), blk, 0, stream>>>(YS, YM);

    // 7) LN -> *z1 -> LN
    ln_gate_ln_kernel<<<dim3(POSB / 256), blk, 0, stream>>>(
        YM, Z1, ong, onb, n1g, n1b, Y2);

    // 8) out_proj: out[b] = Wop(128x128) * Y2[b](128x4096)
    gemm_f32_wmma<<<dim3(LL / 256, DD / 32, BSZ), blk, 0, stream>>>(
        Wop, Y2, out, DD, DD, LL,
        /*aDiv=*/1, /*strideA=*/0, (long long)DD * LL, (long long)DD * LL);
}